// NetSEIR_28930899706602
// MI455X (gfx1250) — compile-verified
//
#include <hip/hip_runtime.h>
#include <hip/hip_bf16.h>

// ---------------------------------------------------------------------------
// NetSEIR: phase 1 = 3 tiny MLP rate nets via V_WMMA_F32_16X16X4_F32 (fp32
// WMMA keeps exact precision); phase 2 = inherently-serial nonlinear SEIR
// scan on one lane, fed by interleaved float4 rate records and emitting
// 16B-aligned b128 trajectory stores.
// ---------------------------------------------------------------------------

#define T_STEPS 1000000
#define N_TILES (T_STEPS / 16)        // 62500 16-row tiles
#define WAVES_PER_BLOCK 8             // 256 threads = 8 wave32

typedef __attribute__((ext_vector_type(2))) float v2f;
typedef __attribute__((ext_vector_type(8))) float v8f;

__device__ __forceinline__ float sigmoidf(float x) {
    return 1.0f / (1.0f + __expf(-x));
}

// Phase 1: rates[t*4+0]=beta, +1=gamma, +2=sigma (padded float4 records).
// One wave computes one 16-row tile. B packs w_beta in cols 0..7 and
// w_gamma in cols 8..15; a second accumulator carries w_sigma in cols 0..7.
__global__ __launch_bounds__(256)
void seir_rates_wmma(const float* __restrict__ X,
                     const float* __restrict__ wb, const float* __restrict__ wb1,
                     const float* __restrict__ wg, const float* __restrict__ wg1,
                     const float* __restrict__ wsg, const float* __restrict__ ws1,
                     float* __restrict__ rates)
{
    const int lane = threadIdx.x & 31;
    const int wave = threadIdx.x >> 5;
    const int tile = blockIdx.x * WAVES_PER_BLOCK + wave;
    if (tile >= N_TILES) return;            // wave-uniform guard (EXEC stays full)

    const int half = lane >> 4;             // which 16-lane half
    const int n    = lane & 15;             // column 0..15 (also A-row selector)
    const int row  = tile * 16 + n;         // X row held by this lane (A matrix)

    // ---- Branchless, hoisted B-matrix construction (loop-invariant) ----
    const int   nc    = n & 7;
    const float* wc   = (n < 8) ? wb : wg;          // pointer select, no exec split
    const float smask = (n < 8) ? 1.0f : 0.0f;      // sigma occupies cols 0..7 only

    float bC[8], bS[8];
    #pragma unroll
    for (int kb = 0; kb < 4; ++kb) {
        const int k0 = 4 * kb + 2 * half;           // K row of this lane's .x slot
        bC[2 * kb]     = wc[k0 * 8 + nc];
        bC[2 * kb + 1] = wc[(k0 + 1) * 8 + nc];
        bS[2 * kb]     = wsg[k0 * 8 + nc] * smask;
        bS[2 * kb + 1] = wsg[(k0 + 1) * 8 + nc] * smask;
    }

    v8f accBG = {0.f, 0.f, 0.f, 0.f, 0.f, 0.f, 0.f, 0.f};   // beta|gamma
    v8f accS  = {0.f, 0.f, 0.f, 0.f, 0.f, 0.f, 0.f, 0.f};   // sigma

    const float* __restrict__ xrow = X + row * 16;

    #pragma unroll
    for (int kb = 0; kb < 4; ++kb) {
        const int k0 = 4 * kb + 2 * half;
        // A: 16x4 fp32 slice of X (ISA 7.12.2) — one b64 load per K-block.
        v2f a = *reinterpret_cast<const v2f*>(xrow + k0);
        v2f bBG = {bC[2 * kb], bC[2 * kb + 1]};
        v2f bS2 = {bS[2 * kb], bS[2 * kb + 1]};

        accBG = __builtin_amdgcn_wmma_f32_16x16x4_f32(
                    false, a, false, bBG, (short)0, accBG, false, false);
        accS  = __builtin_amdgcn_wmma_f32_16x16x4_f32(
                    false, a, false, bS2, (short)0, accS,  false, false);
    }

    // Epilogue: relu, *w1, 8-lane xor-tree reduction over columns, sigmoid.
    const float* w1p = (n < 8) ? wb1 : wg1;
    const float w1c  = w1p[nc];                     // beta or gamma head
    const float w1s  = ws1[nc] * smask;             // sigma head (cols 0..7)

    #pragma unroll
    for (int i = 0; i < 8; ++i) {
        float hBG = fmaxf(accBG[i], 0.f) * w1c;
        float hS  = fmaxf(accS[i],  0.f) * w1s;

        hBG += __shfl_xor(hBG, 1); hBG += __shfl_xor(hBG, 2); hBG += __shfl_xor(hBG, 4);
        hS  += __shfl_xor(hS,  1); hS  += __shfl_xor(hS,  2); hS  += __shfl_xor(hS,  4);

        if ((lane & 7) == 0) {
            const int m = i + 8 * half;                  // row inside tile
            const int t = tile * 16 + m;                 // global time step
            const int gcol = (lane >> 3) & 1;            // 0: beta/sigma cols, 1: gamma cols
            rates[t * 4 + gcol] = sigmoidf(hBG);         // beta -> +0, gamma -> +1
            if (gcol == 0) {
                rates[t * 4 + 2] = sigmoidf(hS);         // sigma -> +2
            }
        }
    }
}

// One SEIR step; updates state in place.
__device__ __forceinline__ void seir_step(const float4 r, const float invN,
                                          float& S, float& E, float& I,
                                          float& Ic, float& R)
{
    const float b = r.x, g = r.y, s = r.z;
    const float StoE = b * Ic * S * invN;
    const float EtoI = s * E;
    const float ItoR = g * Ic;
    const float nS  = S - StoE;
    const float nE  = E + StoE - EtoI;
    const float nI  = EtoI;
    const float nIc = I + EtoI - ItoR;   // uses OLD I (3rd element), per reference
    const float nR  = R + ItoR;
    S = nS; E = nE; I = nI; Ic = nIc; R = nR;
}

// Phase 2: the nonlinear SEIR recurrence — one serial dependency chain.
// Peels 3 steps so the remaining 999996 steps emit 16B-aligned b128 stores
// (row stride T_STEPS is a multiple of 4, trajectory index starts at 4).
__global__ void seir_scan(const float* __restrict__ rates,
                          const float* __restrict__ init_state,
                          const float* __restrict__ Nptr,
                          float* __restrict__ out)
{
    if (threadIdx.x != 0 || blockIdx.x != 0) return;

    float S  = init_state[0];
    float E  = init_state[1];
    float I  = init_state[2];
    float Ic = init_state[3];
    float R  = init_state[4];
    const float invN = 1.0f / Nptr[0];

    out[0 * T_STEPS] = S;
    out[1 * T_STEPS] = E;
    out[2 * T_STEPS] = I;
    out[3 * T_STEPS] = Ic;
    out[4 * T_STEPS] = R;

    const float4* __restrict__ r4 = reinterpret_cast<const float4*>(rates);

    // Peel t = 0..2 (writes trajectory indices 1..3).
    #pragma unroll
    for (int t = 0; t < 3; ++t) {
        seir_step(r4[t], invN, S, E, I, Ic, R);
        out[0 * T_STEPS + 1 + t] = S;
        out[1 * T_STEPS + 1 + t] = E;
        out[2 * T_STEPS + 1 + t] = I;
        out[3 * T_STEPS + 1 + t] = Ic;
        out[4 * T_STEPS + 1 + t] = R;
    }

    // Main: t = 3 .. 999998 in groups of 4; stores land at index 4 + 4*grp.
    for (int grp = 0; grp < (T_STEPS - 4) / 4; ++grp) {
        const int tbase = 3 + grp * 4;
        float aS[4], aE[4], aI[4], aIc[4], aR[4];
        #pragma unroll
        for (int j = 0; j < 4; ++j) {
            seir_step(r4[tbase + j], invN, S, E, I, Ic, R);
            aS[j] = S; aE[j] = E; aI[j] = I; aIc[j] = Ic; aR[j] = R;
        }
        const int o = 1 + tbase;   // = 4 + 4*grp, 16B aligned
        *reinterpret_cast<float4*>(out + 0 * T_STEPS + o) = make_float4(aS[0], aS[1], aS[2], aS[3]);
        *reinterpret_cast<float4*>(out + 1 * T_STEPS + o) = make_float4(aE[0], aE[1], aE[2], aE[3]);
        *reinterpret_cast<float4*>(out + 2 * T_STEPS + o) = make_float4(aI[0], aI[1], aI[2], aI[3]);
        *reinterpret_cast<float4*>(out + 3 * T_STEPS + o) = make_float4(aIc[0], aIc[1], aIc[2], aIc[3]);
        *reinterpret_cast<float4*>(out + 4 * T_STEPS + o) = make_float4(aR[0], aR[1], aR[2], aR[3]);
    }
}

extern "C" void kernel_launch(void* const* d_in, const int* in_sizes, int n_in,
                              void* d_out, int out_size, void* d_ws, size_t ws_size,
                              hipStream_t stream) {
    // setup_inputs() order:
    const float* X    = (const float*)d_in[0];   // [T,16]
    const float* wb   = (const float*)d_in[1];   // [16,8]
    const float* wb1  = (const float*)d_in[2];   // [8]
    const float* wg   = (const float*)d_in[3];   // [16,8]
    const float* wg1  = (const float*)d_in[4];   // [8]
    const float* wsg  = (const float*)d_in[5];   // [16,8]
    const float* ws1  = (const float*)d_in[6];   // [8]
    const float* init = (const float*)d_in[7];   // [5]
    const float* Np   = (const float*)d_in[8];   // [1]

    float* out   = (float*)d_out;                // [5, T]
    float* rates = (float*)d_ws;                 // [T, 4] interleaved b,g,s,pad

    const int blocks = (N_TILES + WAVES_PER_BLOCK - 1) / WAVES_PER_BLOCK;
    seir_rates_wmma<<<blocks, 256, 0, stream>>>(X, wb, wb1, wg, wg1, wsg, ws1, rates);
    seir_scan<<<1, 32, 0, stream>>>(rates, init, Np, out);
}